// SASNorm_29669634081224
// MI455X (gfx1250) — compile-verified
//
#include <hip/hip_runtime.h>
#include <hip/hip_bf16.h>

#define EPS_F      1e-8f
#define SIGMA_MINF 1e-4f

constexpr int B_  = 32;
constexpr int T_  = 8192;
constexpr int C_  = 64;

typedef __attribute__((ext_vector_type(2))) float v2f;
typedef __attribute__((ext_vector_type(8))) float v8f;

// ---------------- workspace layout (floats) ----------------
// [0      , 2048)  sum_mask  (B*C)
// [2048   , 4096)  sum_x
// [4096   , 6144)  sum_x2
// [6144   , 6176)  gap_ratio (B)
// [6176   , 8224)  mu        (B*C)
// [8224   , 10272) inv_sigma (B*C)

__global__ void zero_ws_kernel(float* ws, int n) {
    int i = blockIdx.x * blockDim.x + threadIdx.x;
    if (i < n) ws[i] = 0.0f;
}

// ---------------- pass 1: T-axis reductions via f32 WMMA ----------------
// grid.x = B * (C/16) = 128 ; grid.y = TSPLIT = 8 ; block = 256 (8 waves)
// Each WMMA: D(16x16) = A(16ch x 4t) * ones(4x16) + C  -> row sums accumulate
// in f32 exactly. Lane l supplies A[m = l&15, k = 2*(l>>4)+v].
__global__ void reduce_wmma_kernel(const float* __restrict__ x,
                                   const float* __restrict__ mask,
                                   float* __restrict__ sum_m,
                                   float* __restrict__ sum_x,
                                   float* __restrict__ sum_x2) {
    const int bc   = blockIdx.x;          // 0..127
    const int b    = bc >> 2;
    const int c0   = (bc & 3) * 16;
    const int tbase = blockIdx.y * (T_ / 8);   // 1024-wide T slice

    const int lane  = threadIdx.x & 31;
    const int wave  = threadIdx.x >> 5;   // 0..7
    const int m     = lane & 15;          // channel row within 16-group
    const int khalf = lane >> 4;          // 0 or 1 -> K base 0 or 2

    const size_t baseBC = ((size_t)b * T_) * C_ + (size_t)(c0 + m);

    v8f accm  = {};
    v8f accx  = {};
    v8f accx2 = {};
    v2f bones; bones[0] = 1.0f; bones[1] = 1.0f;   // B = all-ones 4x16

    #pragma unroll 4
    for (int i = 0; i < 32; ++i) {        // 8 waves * 32 iters * 4 t = 1024 t
        const int t0 = tbase + (wave + i * 8) * 4;
        const size_t a0 = baseBC + (size_t)(t0 + 2 * khalf) * C_;
        const size_t a1 = a0 + C_;
        v2f ax, am;
        ax[0] = x[a0];    ax[1] = x[a1];
        am[0] = mask[a0]; am[1] = mask[a1];
        v2f ax2 = ax * ax;
        accm  = __builtin_amdgcn_wmma_f32_16x16x4_f32(false, am,  false, bones,
                                                      (short)0, accm,  false, false);
        accx  = __builtin_amdgcn_wmma_f32_16x16x4_f32(false, ax,  false, bones,
                                                      (short)0, accx,  false, false);
        accx2 = __builtin_amdgcn_wmma_f32_16x16x4_f32(false, ax2, false, bones,
                                                      (short)0, accx2, false, false);
    }

    // D layout: VGPR r -> lanes 0-15: M=r (N=lane) ; lanes 16-31: M=8+r.
    // Column N==0 lives in lanes 0 and 16; each holds 8 row-sums.
    if ((lane & 15) == 0) {
        const int crow = c0 + 8 * khalf;
        #pragma unroll
        for (int v = 0; v < 8; ++v) {
            const int cidx = b * C_ + crow + v;
            atomicAdd(&sum_m [cidx], accm [v]);
            atomicAdd(&sum_x [cidx], accx [v]);
            atomicAdd(&sum_x2[cidx], accx2[v]);
        }
    }
}

// ---------------- pass 1b: longest missing run on channel 0 ----------------
// One wave per b. Lane l scans a 256-step segment; lane 0 merges.
__global__ void gap_kernel(const float* __restrict__ mask, float* __restrict__ gapr) {
    const int b    = blockIdx.x;
    const int lane = threadIdx.x;         // blockDim = 32
    const int SEG  = T_ / 32;             // 256
    const size_t base = ((size_t)b * T_ + (size_t)lane * SEG) * C_;

    int pref = 0, mx = 0, run = 0;
    bool in_pref = true;
    for (int i = 0; i < SEG; ++i) {
        const bool miss = mask[base + (size_t)i * C_] < 0.5f;
        if (miss) {
            ++run;
            if (in_pref) ++pref;
        } else {
            if (run > mx) mx = run;
            run = 0;
            in_pref = false;
        }
    }
    if (run > mx) mx = run;
    const int  suff = run;
    const bool full = in_pref;            // whole segment missing

    __shared__ int s_pref[32], s_suff[32], s_mx[32], s_full[32];
    s_pref[lane] = pref; s_suff[lane] = suff; s_mx[lane] = mx; s_full[lane] = full ? 1 : 0;
    __syncthreads();

    if (lane == 0) {
        int gmx = 0, gsuff = 0;
        for (int i = 0; i < 32; ++i) {
            const int cand = gsuff + s_pref[i];    // run crossing boundary
            if (cand > gmx)    gmx = cand;
            if (s_mx[i] > gmx) gmx = s_mx[i];
            gsuff = s_full[i] ? (gsuff + SEG) : s_suff[i];
        }
        if (gsuff > gmx) gmx = gsuff;
        gapr[b] = (float)gmx / (float)T_;
    }
}

// ---------------- pass 2: per-(b,c) stats ----------------
__device__ __forceinline__ float softplus_f(float v) {
    return (v > 20.0f) ? v : logf(1.0f + expf(v));
}

__global__ void finalize_kernel(const float* __restrict__ sum_m,
                                const float* __restrict__ sum_x,
                                const float* __restrict__ sum_x2,
                                const float* __restrict__ gapr,
                                const int*   __restrict__ phase_id,
                                const float* __restrict__ anchor_mu,
                                const float* __restrict__ anchor_ls,
                                const float* __restrict__ tau0_raw,
                                const float* __restrict__ tau1_raw,
                                float* __restrict__ mu_out,
                                float* __restrict__ isg_out) {
    const int idx = blockIdx.x * blockDim.x + threadIdx.x;
    if (idx >= B_ * C_) return;
    const int b = idx >> 6;
    const int c = idx & 63;

    const float S   = sum_m[idx];
    const float msc = fmaxf(S, EPS_F);
    const float sx  = sum_x[idx];
    const float sx2 = sum_x2[idx];

    const float mu_obs = sx / msc;
    float var = (sx2 - 2.0f * mu_obs * sx + mu_obs * mu_obs * S) / msc;
    var = fmaxf(var, 0.0f);
    const float sigma_obs = sqrtf(var + EPS_F);

    const float coverage = S / (float)T_;
    const float tau0 = softplus_f(tau0_raw[0]);
    const float tau1 = softplus_f(tau1_raw[0]);
    const float w = coverage / (coverage + tau0 + tau1 * gapr[b] + EPS_F);

    const int p = phase_id[b];
    const float mu_ref = anchor_mu[p * C_ + c];
    const float ls_ref = anchor_ls[p * C_ + c];

    const float mu = w * mu_obs + (1.0f - w) * mu_ref;
    const float ls = w * logf(sigma_obs + EPS_F) + (1.0f - w) * ls_ref;
    const float sg = fmaxf(expf(ls), SIGMA_MINF);

    mu_out[idx]  = mu;
    isg_out[idx] = 1.0f / sg;
}

// ---------------- pass 3: elementwise apply (L2-hot re-read) ----------------
__global__ void apply_kernel(const float4* __restrict__ x4,
                             const float4* __restrict__ m4,
                             const float*  __restrict__ mu,
                             const float*  __restrict__ isg,
                             const float*  __restrict__ mtok,
                             float4* __restrict__ out4,
                             int total4) {
    const int stride = gridDim.x * blockDim.x;
    for (int idx = blockIdx.x * blockDim.x + threadIdx.x; idx < total4; idx += stride) {
        if (idx + stride < total4) {
            __builtin_prefetch(&x4[idx + stride], 0, 1);   // global_prefetch_b8
        }
        const int    c4   = idx & 15;            // which float4 within the C row
        const size_t row  = (size_t)idx >> 4;    // b*T + t
        const int    b    = (int)(row >> 13);    // T = 8192
        const int    cbase = c4 * 4;
        const int    sidx  = b * C_ + cbase;

        const float4 xv = x4[idx];
        const float4 mv = m4[idx];

        float4 o;
        o.x = mv.x * ((xv.x - mu[sidx + 0]) * isg[sidx + 0]) + (1.0f - mv.x) * mtok[cbase + 0];
        o.y = mv.y * ((xv.y - mu[sidx + 1]) * isg[sidx + 1]) + (1.0f - mv.y) * mtok[cbase + 1];
        o.z = mv.z * ((xv.z - mu[sidx + 2]) * isg[sidx + 2]) + (1.0f - mv.z) * mtok[cbase + 2];
        o.w = mv.w * ((xv.w - mu[sidx + 3]) * isg[sidx + 3]) + (1.0f - mv.w) * mtok[cbase + 3];
        out4[idx] = o;
    }
}

extern "C" void kernel_launch(void* const* d_in, const int* in_sizes, int n_in,
                              void* d_out, int out_size, void* d_ws, size_t ws_size,
                              hipStream_t stream) {
    (void)in_sizes; (void)n_in; (void)out_size; (void)ws_size;

    const float* x         = (const float*)d_in[0];
    const float* mask      = (const float*)d_in[1];
    const int*   phase_id  = (const int*)  d_in[2];
    const float* anchor_mu = (const float*)d_in[3];
    const float* anchor_ls = (const float*)d_in[4];
    const float* tau0_raw  = (const float*)d_in[5];
    const float* tau1_raw  = (const float*)d_in[6];
    const float* mtok      = (const float*)d_in[7];
    float*       out       = (float*)d_out;

    float* ws      = (float*)d_ws;
    float* sum_m   = ws;                 // B*C
    float* sum_x   = ws + 2048;
    float* sum_x2  = ws + 4096;
    float* gapr    = ws + 6144;          // B
    float* s_mu    = ws + 6176;          // B*C
    float* s_isg   = ws + 8224;          // B*C

    // 0) zero the atomic accumulators (ws is poisoned, not re-poisoned between replays)
    zero_ws_kernel<<<(6144 + 255) / 256, 256, 0, stream>>>(ws, 6144);

    // 1) WMMA T-axis reductions + gap scan
    dim3 rgrid(B_ * (C_ / 16), 8);
    reduce_wmma_kernel<<<rgrid, 256, 0, stream>>>(x, mask, sum_m, sum_x, sum_x2);
    gap_kernel<<<B_, 32, 0, stream>>>(mask, gapr);

    // 2) per-(b,c) stats
    finalize_kernel<<<(B_ * C_ + 255) / 256, 256, 0, stream>>>(
        sum_m, sum_x, sum_x2, gapr, phase_id, anchor_mu, anchor_ls,
        tau0_raw, tau1_raw, s_mu, s_isg);

    // 3) elementwise apply
    const int total4 = B_ * T_ * C_ / 4;   // 4,194,304
    apply_kernel<<<4096, 256, 0, stream>>>((const float4*)x, (const float4*)mask,
                                           s_mu, s_isg, mtok, (float4*)out, total4);
}